// LayerNormNoInputBinaryTreeLSTMCell_61366492725518
// MI455X (gfx1250) — compile-verified
//
#include <hip/hip_runtime.h>
#include <hip/hip_bf16.h>
#include <math.h>

// ---------------------------------------------------------------------------
// LayerNorm binary-tree LSTM cell, H=4096.
//   z = W[20480x8192] @ concat(h0,h1)  -> memory-bound GEMV (671 MB of W,
//                                         ~29 us floor at 23.3 TB/s)
//   5x LayerNorm + gates + cell LN     -> tiny pointwise epilogue
// Kernel 1 streams W with NT b128 loads feeding V_WMMA_F32_16X16X4_F32.
// ---------------------------------------------------------------------------

typedef __attribute__((ext_vector_type(2))) float v2f;
typedef __attribute__((ext_vector_type(4))) float v4f;
typedef __attribute__((ext_vector_type(8))) float v8f;

#define HDIM   4096
#define KDIM   8192          // 2*H
#define NROWS  20480         // 5*H
#define BLK_T  256           // 8 waves (wave32)
#define KWAVE  (KDIM / 8)    // 1024 K per wave

// ---------------------------------------------------------------------------
// Kernel 1: z[row] = sum_k W[row,k] * x[k],  x = concat(h0, h1)
// One block = one 16-row tile; 8 waves split K (waves 0-3 -> h0, 4-7 -> h1,
// so B fragments come straight from global, no staging).
//
// WMMA f32 16x16x4: A lane l holds W[row0+(l&15), <2 K-slots>], B lane l
// holds x[<same 2 K-slots>] broadcast across N. Because we construct BOTH
// A and B, the K->slot mapping is a free permutation: a b128 load of
// 4 consecutive floats per lane supplies the fragments of TWO WMMAs
// (slots {0,1} and {2,3}), with the half-wave (l>>4) covering cols +4..+7.
// D: VGPR r, lanes 0-15 -> row r, lanes 16-31 -> row 8+r (cols identical).
// ---------------------------------------------------------------------------
__global__ __launch_bounds__(BLK_T) void gemv_wmma_f32(
    const float* __restrict__ W,
    const float* __restrict__ h0,
    const float* __restrict__ h1,
    float* __restrict__ z)
{
    __shared__ float part[8][16];     // per-wave partial row sums (512 B)

    const int tid  = threadIdx.x;
    const int wave = tid >> 5;
    const int lane = tid & 31;
    const int m    = lane & 15;       // row within tile
    const int half = lane >> 4;       // K sub-block select (0 or 1)

    const size_t row = (size_t)blockIdx.x * 16 + m;
    const float* wp = W + row * (size_t)KDIM + wave * KWAVE + 4 * half;
    const float* xp = ((wave < 4) ? h0 + wave * KWAVE
                                  : h1 + (wave - 4) * KWAVE) + 4 * half;

    v8f acc0 = {}, acc1 = {}, acc2 = {}, acc3 = {};

    // 64 iterations, 4 WMMAs each (256 per wave); 4 accumulators break the
    // C-dependency chain. Each iteration: 2 NT b128 W loads (512 B/instr
    // across the wave) + 2 b128 x loads (merged broadcast, L0-resident).
    for (int k0 = 0; k0 < KWAVE; k0 += 16) {
        v4f a01 = __builtin_nontemporal_load((const v4f*)(wp + k0));
        v4f b01 = *(const v4f*)(xp + k0);
        v4f a23 = __builtin_nontemporal_load((const v4f*)(wp + k0 + 8));
        v4f b23 = *(const v4f*)(xp + k0 + 8);

        v2f a0 = __builtin_shufflevector(a01, a01, 0, 1);
        v2f b0 = __builtin_shufflevector(b01, b01, 0, 1);
        acc0 = __builtin_amdgcn_wmma_f32_16x16x4_f32(false, a0, false, b0,
                                                     (short)0, acc0, false, false);
        v2f a1 = __builtin_shufflevector(a01, a01, 2, 3);
        v2f b1 = __builtin_shufflevector(b01, b01, 2, 3);
        acc1 = __builtin_amdgcn_wmma_f32_16x16x4_f32(false, a1, false, b1,
                                                     (short)0, acc1, false, false);
        v2f a2 = __builtin_shufflevector(a23, a23, 0, 1);
        v2f b2 = __builtin_shufflevector(b23, b23, 0, 1);
        acc2 = __builtin_amdgcn_wmma_f32_16x16x4_f32(false, a2, false, b2,
                                                     (short)0, acc2, false, false);
        v2f a3 = __builtin_shufflevector(a23, a23, 2, 3);
        v2f b3 = __builtin_shufflevector(b23, b23, 2, 3);
        acc3 = __builtin_amdgcn_wmma_f32_16x16x4_f32(false, a3, false, b3,
                                                     (short)0, acc3, false, false);
    }
    v8f acc = (acc0 + acc1) + (acc2 + acc3);

    // Lanes 0 and 16 hold the 16 row results in accumulator VGPRs 0..7.
    if ((lane & 15) == 0) {
        #pragma unroll
        for (int r = 0; r < 8; ++r)
            part[wave][half * 8 + r] = acc[r];
    }
    __syncthreads();

    // Fixed-order sum over the 8 wave partials -> deterministic.
    if (tid < 16) {
        float s = 0.0f;
        #pragma unroll
        for (int w = 0; w < 8; ++w) s += part[w][tid];
        z[(size_t)blockIdx.x * 16 + tid] = s;
    }
}

// ---------------------------------------------------------------------------
// Kernel 2: pointwise epilogue (single block, 1024 threads, 4 elems/thread).
// ---------------------------------------------------------------------------
__device__ __forceinline__ float block_reduce_sum(float v, float* red)
{
    #pragma unroll
    for (int off = 16; off > 0; off >>= 1)
        v += __shfl_down(v, off, 32);
    const int lane = threadIdx.x & 31;
    const int wav  = threadIdx.x >> 5;
    if (lane == 0) red[wav] = v;
    __syncthreads();
    if (wav == 0) {
        float t = red[lane];              // 32 waves
        #pragma unroll
        for (int off = 16; off > 0; off >>= 1)
            t += __shfl_down(t, off, 32);
        if (lane == 0) red[0] = t;
    }
    __syncthreads();
    float r = red[0];
    __syncthreads();
    return r;
}

__device__ __forceinline__ float sigf(float x) { return 1.0f / (1.0f + expf(-x)); }

__global__ __launch_bounds__(1024) void lstm_pointwise(
    const float* __restrict__ z,
    const float* __restrict__ c0,
    const float* __restrict__ c1,
    const float* __restrict__ ffio_g,
    const float* __restrict__ ffio_b,
    const float* __restrict__ u_g,
    const float* __restrict__ u_b,
    const float* __restrict__ c_g,
    const float* __restrict__ c_b,
    float* __restrict__ out)           // [0:H) hidden, [H:2H) cell
{
    __shared__ float red[32];
    const int j0 = threadIdx.x * 4;
    const float invH = 1.0f / (float)HDIM;

    // Stats for the 5 z-parts.
    float mean[5], rstd[5];
    #pragma unroll
    for (int p = 0; p < 5; ++p) {
        v4f v = *(const v4f*)(z + p * HDIM + j0);
        float s  = v[0] + v[1] + v[2] + v[3];
        float ss = v[0]*v[0] + v[1]*v[1] + v[2]*v[2] + v[3]*v[3];
        s  = block_reduce_sum(s, red);
        ss = block_reduce_sum(ss, red);
        float m = s * invH;
        float var = ss * invH - m * m;
        mean[p] = m;
        rstd[p] = rsqrtf(var + 1e-5f);
    }

    // Gates: per-gate LN -> sigmoid (f0,f1,i,o) / tanh (u).
    v4f gate[5];
    #pragma unroll
    for (int p = 0; p < 4; ++p) {
        v4f v  = *(const v4f*)(z + p * HDIM + j0);
        v4f gg = *(const v4f*)(ffio_g + p * HDIM + j0);
        v4f bb = *(const v4f*)(ffio_b + p * HDIM + j0);
        #pragma unroll
        for (int e = 0; e < 4; ++e)
            gate[p][e] = sigf((v[e] - mean[p]) * rstd[p] * gg[e] + bb[e]);
    }
    {
        v4f v  = *(const v4f*)(z + 4 * HDIM + j0);
        v4f gg = *(const v4f*)(u_g + j0);
        v4f bb = *(const v4f*)(u_b + j0);
        #pragma unroll
        for (int e = 0; e < 4; ++e)
            gate[4][e] = tanhf((v[e] - mean[4]) * rstd[4] * gg[e] + bb[e]);
    }

    // Pre-norm cell + its stats.
    v4f c0v = *(const v4f*)(c0 + j0);
    v4f c1v = *(const v4f*)(c1 + j0);
    v4f cp;
    float s = 0.0f, ss = 0.0f;
    #pragma unroll
    for (int e = 0; e < 4; ++e) {
        float c = gate[2][e] * gate[4][e] + gate[0][e] * c0v[e] + gate[1][e] * c1v[e];
        cp[e] = c;
        s += c;
        ss += c * c;
    }
    s  = block_reduce_sum(s, red);
    ss = block_reduce_sum(ss, red);
    float mc = s * invH;
    float rc = rsqrtf(ss * invH - mc * mc + 1e-5f);

    v4f cgv = *(const v4f*)(c_g + j0);
    v4f cbv = *(const v4f*)(c_b + j0);
    v4f cellv, hidv;
    #pragma unroll
    for (int e = 0; e < 4; ++e) {
        float cn = (cp[e] - mc) * rc * cgv[e] + cbv[e];
        cellv[e] = cn;
        hidv[e]  = gate[3][e] * tanhf(cn);
    }
    *(v4f*)(out + j0)        = hidv;
    *(v4f*)(out + HDIM + j0) = cellv;
}

// ---------------------------------------------------------------------------
extern "C" void kernel_launch(void* const* d_in, const int* in_sizes, int n_in,
                              void* d_out, int out_size, void* d_ws, size_t ws_size,
                              hipStream_t stream) {
    const float* h0     = (const float*)d_in[0];
    const float* c0     = (const float*)d_in[1];
    const float* h1     = (const float*)d_in[2];
    const float* c1     = (const float*)d_in[3];
    const float* W      = (const float*)d_in[4];
    const float* ffio_g = (const float*)d_in[5];
    const float* ffio_b = (const float*)d_in[6];
    const float* u_g    = (const float*)d_in[7];
    const float* u_b    = (const float*)d_in[8];
    const float* c_g    = (const float*)d_in[9];
    const float* c_b    = (const float*)d_in[10];

    float* out = (float*)d_out;
    float* z   = (float*)d_ws;               // 20480 floats scratch

    gemv_wmma_f32<<<NROWS / 16, BLK_T, 0, stream>>>(W, h0, h1, z);
    lstm_pointwise<<<1, 1024, 0, stream>>>(z, c0, c1, ffio_g, ffio_b,
                                           u_g, u_b, c_g, c_b, out);
}